// BasicRouter_14018773254407
// MI455X (gfx1250) — compile-verified
//
#include <hip/hip_runtime.h>

typedef __attribute__((ext_vector_type(2))) float v2f;
typedef __attribute__((ext_vector_type(8))) float v8f;

#define N_TOKENS 16384
#define D_IN     2048
#define N_EXP    8

__global__ __launch_bounds__(256) void moe_router_wmma(
    const float* __restrict__ x,   // [16384, 2048]
    const float* __restrict__ Wm,  // [8, 2048]
    const float* __restrict__ bias,// [8]
    float* __restrict__ out)       // logits | weights | idx(i32) | mask(i32)
{
    __shared__ float lds_logits[8][16 * 16];   // 8 waves * 16x16 tile = 8 KB

    const int tid  = threadIdx.x;
    const int wave = tid >> 5;
    const int lane = tid & 31;
    const int m    = lane & 15;     // A: row (token-in-tile); B/C/D: column (expert)
    const int half = lane >> 4;     // selects K pair within the 16x4 / 4x16 fragments

    const int tokBase = (blockIdx.x * 8 + wave) * 16;

    // A: x[tokBase+m][kk + half*2 .. +1]
    const float* aPtr = x + (size_t)(tokBase + m) * D_IN + half * 2;
    // B: W[n][kk + half*2 .. +1], n = column = m; columns 8..15 are zero padding
    const float* bPtr = Wm + (size_t)(m & 7) * D_IN + half * 2;
    const float bscale = (m < N_EXP) ? 1.0f : 0.0f;   // VALU mask, no divergence

    v8f acc = {0.f, 0.f, 0.f, 0.f, 0.f, 0.f, 0.f, 0.f};

#pragma unroll 8
    for (int kk = 0; kk < D_IN; kk += 4) {
        v2f a  = *(const v2f*)(aPtr + kk);
        v2f bb = *(const v2f*)(bPtr + kk);
        bb.x *= bscale;
        bb.y *= bscale;
        // D = A(16x4,f32) * B(4x16,f32) + C  -> v_wmma_f32_16x16x4_f32
        acc = __builtin_amdgcn_wmma_f32_16x16x4_f32(
            /*neg_a=*/false, a, /*neg_b=*/false, bb,
            /*c_mod=*/(short)0, acc, /*reuse_a=*/false, /*reuse_b=*/false);
    }

    // C/D layout: lanes 0-15: VGPR r -> (row=r,     col=lane)
    //             lanes16-31: VGPR r -> (row=r + 8, col=lane-16)
    const float bcol = (m < N_EXP) ? bias[m] : 0.0f;
#pragma unroll
    for (int r = 0; r < 8; ++r) {
        lds_logits[wave][(r + 8 * half) * 16 + m] = acc[r] + bcol;
    }

    __syncthreads();

    // ---- Phase 2: one thread per token (128 tokens per block) ----
    if (tid < 128) {
        const int t   = blockIdx.x * 128 + tid;
        const int w2  = tid >> 4;
        const int row = tid & 15;

        float l[N_EXP];
#pragma unroll
        for (int e = 0; e < N_EXP; ++e) l[e] = lds_logits[w2][row * 16 + e];

        // router_logits
        float4* lo = (float4*)(out + (size_t)t * 8);
        lo[0] = make_float4(l[0], l[1], l[2], l[3]);
        lo[1] = make_float4(l[4], l[5], l[6], l[7]);

        // top-2 (strict '>' keeps lowest index on ties, matching lax.top_k)
        int i0 = 0; float v0 = l[0];
#pragma unroll
        for (int e = 1; e < N_EXP; ++e) { if (l[e] > v0) { v0 = l[e]; i0 = e; } }
        int i1 = -1; float v1 = -3.402823466e+38f;
#pragma unroll
        for (int e = 0; e < N_EXP; ++e) { if (e != i0 && l[e] > v1) { v1 = l[e]; i1 = e; } }

        // normalized top-2 softmax weights: global denominator cancels
        const float e1    = __expf(v1 - v0);
        const float inv   = 1.0f / (1.0f + e1);
        const float w0    = inv;
        const float w1    = e1 * inv;

        float* ew = out + (size_t)N_TOKENS * 8 + (size_t)t * 2;
        ew[0] = w0; ew[1] = w1;

        int* ibase = (int*)(out + (size_t)N_TOKENS * 8 + (size_t)N_TOKENS * 2);
        int* idx   = ibase + (size_t)t * 2;
        idx[0] = i0; idx[1] = i1;

        int* mask = ibase + (size_t)N_TOKENS * 2 + (size_t)t * 16;
        int4* m4  = (int4*)mask;
        m4[0] = make_int4(i0 == 0, i0 == 1, i0 == 2, i0 == 3);
        m4[1] = make_int4(i0 == 4, i0 == 5, i0 == 6, i0 == 7);
        m4[2] = make_int4(i1 == 0, i1 == 1, i1 == 2, i1 == 3);
        m4[3] = make_int4(i1 == 4, i1 == 5, i1 == 6, i1 == 7);
    }
}

extern "C" void kernel_launch(void* const* d_in, const int* in_sizes, int n_in,
                              void* d_out, int out_size, void* d_ws, size_t ws_size,
                              hipStream_t stream) {
    const float* x  = (const float*)d_in[0];
    const float* Wm = (const float*)d_in[1];
    const float* b  = (const float*)d_in[2];
    float* out = (float*)d_out;

    dim3 grid(N_TOKENS / 128);   // 8 waves/block * 16 tokens/wave = 128 tokens/block
    dim3 block(256);
    moe_router_wmma<<<grid, block, 0, stream>>>(x, Wm, b, out);
}